// LinODECell_15135464751357
// MI455X (gfx1250) — compile-verified
//
#include <hip/hip_runtime.h>
#include <hip/hip_bf16.h>

// LinODECell: xhat[b,:] = expm(dt_b * gamma * W) @ x0[b,:]
// Computed as a 30-term Taylor series on the *vector*:
//   y0 = x0;  y_k = (dt_b/k) * (W @ y_{k-1});  xhat = sum_k y_k
// Each term is one [B,64]x[64,64] fp32 GEMM done with V_WMMA_F32_16X16X4_F32.
// ||dt*gamma*W|| <= ~2, so 30 terms converge far below fp32 epsilon.

typedef __attribute__((ext_vector_type(2))) float v2f;
typedef __attribute__((ext_vector_type(8))) float v8f;

#define DMAT   64
#define ROWS   32      // batch rows per block
#define PITCH  68      // LDS row pitch in floats: 272B (%8==0, bank-conflict-free)
#define NTERMS 30      // 2^30/30! ~ 4e-24  << fp32 eps

__global__ __launch_bounds__(256)
void linode_expm_action_kernel(const float* __restrict__ dt,
                               const float* __restrict__ x0,
                               const float* __restrict__ scal,
                               const float* __restrict__ weight,
                               float* __restrict__ out)
{
    __shared__ float Wl[DMAT * PITCH];       // gamma * W, row-major, padded
    __shared__ float Yl[2][ROWS * PITCH];    // ping-pong state tile

    const int tid  = threadIdx.x;
    const int wave = tid >> 5;               // 0..7
    const int lane = tid & 31;
    const int half = lane >> 4;              // 0/1 (lane group)
    const int ln   = lane & 15;
    const int tm   = wave >> 2;              // row-tile 0..1 (16 rows each)
    const int tn   = wave & 3;               // col-tile 0..3 (16 cols each)
    const int r0   = blockIdx.x * ROWS;      // first batch row of this block

    // ---- stage gamma*W into LDS ----
    const float gamma = scal[0];
    for (int i = tid; i < DMAT * DMAT; i += 256) {
        int r = i >> 6, c = i & 63;
        Wl[r * PITCH + c] = gamma * weight[i];
    }

    // ---- per-lane constants: dt for the 8 output rows this lane owns ----
    const int rowBase = tm * 16 + half * 8;          // row offset within tile
    float dtv[8];
    #pragma unroll
    for (int v = 0; v < 8; ++v)
        dtv[v] = dt[r0 + rowBase + v];

    // ---- S = x0 in C/D fragment layout (registers); Y[0] = x0 tile (LDS) ----
    v8f S;
    #pragma unroll
    for (int v = 0; v < 8; ++v)
        S[v] = x0[(size_t)(r0 + rowBase + v) * DMAT + tn * 16 + ln];
    for (int i = tid; i < ROWS * DMAT; i += 256) {
        int r = i >> 6, c = i & 63;
        Yl[0][r * PITCH + c] = x0[(size_t)(r0 + r) * DMAT + c];
    }
    __syncthreads();

    // ---- preload B fragments into registers (constant across all terms) ----
    // D[r][c] = sum_j A[r][j] * B[j][c] with B[j][c] = W[c][j]  (Y @ W^T)
    // B frag per ISA layout: N = lane%16, K = 2*half + v
    v2f bf[16];
    #pragma unroll
    for (int kb = 0; kb < 16; ++kb)
        bf[kb] = *(const v2f*)&Wl[(tn * 16 + ln) * PITCH + kb * 4 + 2 * half];

    // Per-wave LDS addresses for A-fragment reads / Y-tile writes (both buffers).
    const float* arow0 = &Yl[0][(tm * 16 + ln) * PITCH + 2 * half];
    const float* arow1 = &Yl[1][(tm * 16 + ln) * PITCH + 2 * half];
    float* wrow0 = &Yl[0][rowBase * PITCH + tn * 16 + ln];
    float* wrow1 = &Yl[1][rowBase * PITCH + tn * 16 + ln];

    // ---- fully unrolled Taylor recurrence: 1/k folds to literal constants ----
    #pragma unroll
    for (int k = 1; k <= NTERMS; ++k) {
        const float invk = 1.0f / (float)k;            // compile-time constant
        const float* ybase = (k & 1) ? arow0 : arow1;  // read Y[(k-1)%2]
        float*       wbase = (k & 1) ? wrow1 : wrow0;  // write Y[k%2]

        // A frag per ISA layout: M = lane%16, K = 2*half + v  -> ds_load_b64 x2
        v8f acc = {};
        #pragma unroll
        for (int kb = 0; kb < 16; ++kb) {
            v2f af = *(const v2f*)(ybase + kb * 4);
            acc = __builtin_amdgcn_wmma_f32_16x16x4_f32(
                false, af, false, bf[kb], (short)0, acc, false, false);
        }

        // scale by dt/k, accumulate into S, write next Y tile (ping-pong).
        // One barrier per term: reads hit Y[(k-1)%2], writes hit Y[k%2].
        #pragma unroll
        for (int v = 0; v < 8; ++v) {
            float val = acc[v] * (dtv[v] * invk);
            wbase[v * PITCH] = val;
            S[v] += val;
        }
        __syncthreads();
    }

    // ---- write xhat ----
    #pragma unroll
    for (int v = 0; v < 8; ++v)
        out[(size_t)(r0 + rowBase + v) * DMAT + tn * 16 + ln] = S[v];
}

extern "C" void kernel_launch(void* const* d_in, const int* in_sizes, int n_in,
                              void* d_out, int out_size, void* d_ws, size_t ws_size,
                              hipStream_t stream) {
    const float* dt     = (const float*)d_in[0];   // [B]
    const float* x0     = (const float*)d_in[1];   // [B,64]
    const float* scalar = (const float*)d_in[2];   // [1]
    const float* weight = (const float*)d_in[3];   // [64,64]
    float* out          = (float*)d_out;           // [B,64]

    const int B = in_sizes[0];                     // 8192 (multiple of ROWS)
    const int nblocks = B / ROWS;

    linode_expm_action_kernel<<<nblocks, 256, 0, stream>>>(dt, x0, scalar, weight, out);
}